// DiffusionConv_90520730730511
// MI455X (gfx1250) — compile-verified
//
#include <hip/hip_runtime.h>

#define DIN  128
#define DOUT 128

typedef float v2f __attribute__((ext_vector_type(2)));
typedef float v8f __attribute__((ext_vector_type(8)));

// ---------------------------------------------------------------------------
// 1) init: deg = 1 (self-loop), CSR cursors = 0
// ---------------------------------------------------------------------------
__global__ void init_kernel(int* __restrict__ degf, int* __restrict__ degb,
                            int* __restrict__ curf, int* __restrict__ curb, int n) {
    int i = blockIdx.x * blockDim.x + threadIdx.x;
    if (i < n) { degf[i] = 1; degb[i] = 1; curf[i] = 0; curb[i] = 0; }
}

// ---------------------------------------------------------------------------
// 2) integer degree histogram: fwd conv aggregates on dst, bwd conv on src
// ---------------------------------------------------------------------------
__global__ void count_deg_kernel(const int* __restrict__ src, const int* __restrict__ dst,
                                 int* __restrict__ degf, int* __restrict__ degb, int e) {
    int i = blockIdx.x * blockDim.x + threadIdx.x;
    if (i < e) {
        atomicAdd(&degf[dst[i]], 1);
        atomicAdd(&degb[src[i]], 1);
    }
}

// ---------------------------------------------------------------------------
// 3) dinv = rsqrt(deg)  (degrees preserved; needed for CSR bucket sizes)
// ---------------------------------------------------------------------------
__global__ void dinv_kernel(const int* __restrict__ degf, const int* __restrict__ degb,
                            float* __restrict__ df, float* __restrict__ db, int n) {
    int i = blockIdx.x * blockDim.x + threadIdx.x;
    if (i < n) {
        df[i] = rsqrtf((float)degf[i]);
        db[i] = rsqrtf((float)degb[i]);
    }
}

// ---------------------------------------------------------------------------
// 4) Exclusive scan of (deg-1) -> CSR offsets. Single workgroup, LDS
//    Hillis-Steele per 1024-chunk with running carry. Deterministic.
// ---------------------------------------------------------------------------
__global__ void __launch_bounds__(1024)
scan_kernel(const int* __restrict__ degf, const int* __restrict__ degb,
            int* __restrict__ offf, int* __restrict__ offb, int n) {
    __shared__ int sf[1024];
    __shared__ int sb[1024];
    int carryf = 0, carryb = 0;
    const int t = threadIdx.x;
    for (int base = 0; base < n; base += 1024) {
        const int i = base + t;
        const int vf = (i < n) ? (degf[i] - 1) : 0;   // edge count excl. self-loop
        const int vb = (i < n) ? (degb[i] - 1) : 0;
        sf[t] = vf;
        sb[t] = vb;
        __syncthreads();
        for (int d = 1; d < 1024; d <<= 1) {
            int tf = 0, tb = 0;
            if (t >= d) { tf = sf[t - d]; tb = sb[t - d]; }
            __syncthreads();
            if (t >= d) { sf[t] += tf; sb[t] += tb; }
            __syncthreads();
        }
        if (i < n) {
            offf[i] = carryf + sf[t] - vf;            // exclusive prefix
            offb[i] = carryb + sb[t] - vb;
        }
        carryf += sf[1023];                            // chunk totals (uniform)
        carryb += sb[1023];
        __syncthreads();                               // protect sf/sb before reuse
    }
}

// ---------------------------------------------------------------------------
// 5) CSR fill: idxf = src ids bucketed by dst; idxb = dst ids bucketed by src
// ---------------------------------------------------------------------------
__global__ void fill_csr_kernel(const int* __restrict__ src, const int* __restrict__ dst,
                                const int* __restrict__ offf, const int* __restrict__ offb,
                                int* __restrict__ curf, int* __restrict__ curb,
                                int* __restrict__ idxf, int* __restrict__ idxb, int e) {
    int i = blockIdx.x * blockDim.x + threadIdx.x;
    if (i < e) {
        const int s = src[i];
        const int d = dst[i];
        const int pf = atomicAdd(&curf[d], 1);
        idxf[offf[d] + pf] = s;
        const int pb = atomicAdd(&curb[s], 1);
        idxb[offb[s] + pb] = d;
    }
}

// ---------------------------------------------------------------------------
// 6) Dual GEMM with fp32 WMMA: Hf = X*Wf, Hb = X*Wb.  (unchanged from R1;
//    proven to lower to v_wmma_f32_16x16x4_f32)
// ---------------------------------------------------------------------------
__global__ void __launch_bounds__(256)
gemm_dual_wmma_kernel(const float* __restrict__ X,
                      const float* __restrict__ Wf,
                      const float* __restrict__ Wb,
                      float* __restrict__ Hf,
                      float* __restrict__ Hb) {
    const int wave = threadIdx.x >> 5;   // 0..7
    const int lane = threadIdx.x & 31;
    const int row0 = blockIdx.x * 16;
    const int col0 = wave * 16;
    const int m    = lane & 15;
    const int hi   = lane >> 4;

    v8f cf = {};
    v8f cb = {};

    const float* xrow = X + (size_t)(row0 + m) * DIN;

#pragma unroll 8
    for (int k = 0; k < DIN; k += 4) {
        const int kk = k + hi * 2;
        v2f a = *(const v2f*)(xrow + kk);
        v2f bfr, bbr;
        bfr.x = Wf[(size_t)kk * DOUT + col0 + m];
        bfr.y = Wf[(size_t)(kk + 1) * DOUT + col0 + m];
        bbr.x = Wb[(size_t)kk * DOUT + col0 + m];
        bbr.y = Wb[(size_t)(kk + 1) * DOUT + col0 + m];
        cf = __builtin_amdgcn_wmma_f32_16x16x4_f32(false, a, false, bfr, (short)0, cf, false, false);
        cb = __builtin_amdgcn_wmma_f32_16x16x4_f32(false, a, false, bbr, (short)0, cb, false, false);
    }

#pragma unroll
    for (int j = 0; j < 8; ++j) {
        const int r = row0 + hi * 8 + j;
        Hf[(size_t)r * DOUT + col0 + m] = cf[j];
        Hb[(size_t)r * DOUT + col0 + m] = cb[j];
    }
}

// ---------------------------------------------------------------------------
// 7) Gather + finalize: one wave per node, 4 channels per lane.
//    out[i] = relu( bf + bb + dinvf[i]^2*Hf[i] + dinvb[i]^2*Hb[i]
//                   + sum_{s in fwdCSR[i]} dinvf[s]*dinvf[i]*Hf[s]
//                   + sum_{d in bwdCSR[i]} dinvb[d]*dinvb[i]*Hb[d] )
//    Atomic-free; fully coalesced 512B row reads, all L2-resident.
// ---------------------------------------------------------------------------
__global__ void __launch_bounds__(256)
gather_kernel(const float* __restrict__ Hf, const float* __restrict__ Hb,
              const int* __restrict__ idxf, const int* __restrict__ idxb,
              const int* __restrict__ offf, const int* __restrict__ offb,
              const int* __restrict__ degf, const int* __restrict__ degb,
              const float* __restrict__ df, const float* __restrict__ db,
              const float* __restrict__ bf, const float* __restrict__ bb,
              float* __restrict__ out, int n) {
    const int node = blockIdx.x * 8 + (threadIdx.x >> 5);
    if (node >= n) return;
    const int lane = threadIdx.x & 31;
    const int c = lane * 4;

    const float dfi = df[node];
    const float dbi = db[node];

    const float4 biasf = *(const float4*)(bf + c);
    const float4 biasb = *(const float4*)(bb + c);
    const float4 hfi   = *(const float4*)(Hf + (size_t)node * DOUT + c);
    const float4 hbi   = *(const float4*)(Hb + (size_t)node * DOUT + c);

    float ax = biasf.x + biasb.x + dfi * dfi * hfi.x + dbi * dbi * hbi.x;
    float ay = biasf.y + biasb.y + dfi * dfi * hfi.y + dbi * dbi * hbi.y;
    float az = biasf.z + biasb.z + dfi * dfi * hfi.z + dbi * dbi * hbi.z;
    float aw = biasf.w + biasb.w + dfi * dfi * hfi.w + dbi * dbi * hbi.w;

    // forward conv: messages from src neighbors (edges whose dst == node)
    {
        const int s0 = offf[node];
        const int cnt = degf[node] - 1;
        for (int j = s0; j < s0 + cnt; ++j) {
            const int s = idxf[j];
            const float nf = dfi * df[s];
            const float4 v = *(const float4*)(Hf + (size_t)s * DOUT + c);
            ax += nf * v.x; ay += nf * v.y; az += nf * v.z; aw += nf * v.w;
        }
    }
    // backward conv: messages from dst neighbors (edges whose src == node)
    {
        const int s0 = offb[node];
        const int cnt = degb[node] - 1;
        for (int j = s0; j < s0 + cnt; ++j) {
            const int d = idxb[j];
            const float nb = dbi * db[d];
            const float4 v = *(const float4*)(Hb + (size_t)d * DOUT + c);
            ax += nb * v.x; ay += nb * v.y; az += nb * v.z; aw += nb * v.w;
        }
    }

    float4 r;
    r.x = fmaxf(ax, 0.0f);
    r.y = fmaxf(ay, 0.0f);
    r.z = fmaxf(az, 0.0f);
    r.w = fmaxf(aw, 0.0f);
    *(float4*)(out + (size_t)node * DOUT + c) = r;
}

// ---------------------------------------------------------------------------
// Launch
// ---------------------------------------------------------------------------
extern "C" void kernel_launch(void* const* d_in, const int* in_sizes, int n_in,
                              void* d_out, int out_size, void* d_ws, size_t ws_size,
                              hipStream_t stream) {
    const float* x    = (const float*)d_in[0];
    const int*   eidx = (const int*)d_in[1];     // [2, E]: row0 = src, row1 = dst
    const float* Wf   = (const float*)d_in[2];
    const float* bf   = (const float*)d_in[3];
    const float* Wb   = (const float*)d_in[4];
    const float* bb   = (const float*)d_in[5];

    const int N = in_sizes[0] / DIN;             // 50000
    const int E = in_sizes[1] / 2;               // 625000
    const int* src = eidx;
    const int* dst = eidx + E;

    float* out = (float*)d_out;

    // Workspace layout
    float* Hf   = (float*)d_ws;                  // N*128 f32
    float* Hb   = Hf + (size_t)N * DOUT;         // N*128 f32
    float* dinf = Hb + (size_t)N * DOUT;         // N f32
    float* dinb = dinf + N;                      // N f32
    int*   degf = (int*)(dinb + N);              // N i32
    int*   degb = degf + N;                      // N i32
    int*   offf = degb + N;                      // N i32
    int*   offb = offf + N;                      // N i32
    int*   curf = offb + N;                      // N i32
    int*   curb = curf + N;                      // N i32
    int*   idxf = curb + N;                      // E i32
    int*   idxb = idxf + E;                      // E i32

    init_kernel<<<(N + 255) / 256, 256, 0, stream>>>(degf, degb, curf, curb, N);
    count_deg_kernel<<<(E + 255) / 256, 256, 0, stream>>>(src, dst, degf, degb, E);
    dinv_kernel<<<(N + 255) / 256, 256, 0, stream>>>(degf, degb, dinf, dinb, N);
    scan_kernel<<<1, 1024, 0, stream>>>(degf, degb, offf, offb, N);
    fill_csr_kernel<<<(E + 255) / 256, 256, 0, stream>>>(src, dst, offf, offb,
                                                          curf, curb, idxf, idxb, E);

    // dual GEMM via fp32 WMMA (N divisible by 16: 50000 = 3125*16)
    gemm_dual_wmma_kernel<<<N / 16, 256, 0, stream>>>(x, Wf, Wb, Hf, Hb);

    // atomic-free gather + bias + self-loop + ReLU (one wave per node)
    gather_kernel<<<(N + 7) / 8, 256, 0, stream>>>(Hf, Hb, idxf, idxb, offf, offb,
                                                    degf, degb, dinf, dinb, bf, bb, out, N);
}